// GNNClassifier_37245956391181
// MI455X (gfx1250) — compile-verified
//
#include <hip/hip_runtime.h>
#include <hip/hip_bf16.h>

typedef __attribute__((ext_vector_type(2))) float v2f;
typedef __attribute__((ext_vector_type(8))) float v8f;

// ---------------------------------------------------------------------------
// Generic fill
// ---------------------------------------------------------------------------
__global__ void k_fill_f32(float* __restrict__ p, float v, long n) {
    long i = (long)blockIdx.x * blockDim.x + threadIdx.x;
    long stride = (long)gridDim.x * blockDim.x;
    for (; i < n; i += stride) p[i] = v;
}

// ---------------------------------------------------------------------------
// Degree accumulation: deg[dst[e]] += 1  (deg pre-filled with 1.0 = self loop)
// ---------------------------------------------------------------------------
__global__ void k_degree(const int* __restrict__ dst, float* __restrict__ deg, int E) {
    int e = blockIdx.x * blockDim.x + threadIdx.x;
    if (e >= E) return;
    atomicAdd(&deg[dst[e]], 1.0f);
}

__global__ void k_rsqrt(const float* __restrict__ deg, float* __restrict__ dis, int N) {
    int i = blockIdx.x * blockDim.x + threadIdx.x;
    if (i >= N) return;
    dis[i] = rsqrtf(deg[i]);
}

// ---------------------------------------------------------------------------
// H = X (M x 128) @ W (128 x 128) using V_WMMA_F32_16X16X4_F32.
// One wave per 16-row strip; 8 column tiles (N=128) accumulated in registers.
// A-frag layout: lanes 0-15 -> M=lane, K = {k,k+1}; lanes 16-31 -> M=lane-16,
//                K = {k+2,k+3}.
// B-frag layout: v.x/v.y hold rows K={k,k+1} (lanes 0-15) / K={k+2,k+3}
//                (lanes 16-31), N = lane&15.
// D tile: VGPR r -> row r (lanes 0-15) / row r+8 (lanes 16-31), col = lane&15.
// ---------------------------------------------------------------------------
__global__ void k_gemm_wmma_f32(const float* __restrict__ X,
                                const float* __restrict__ W,
                                float* __restrict__ H, int M) {
    int wave = (blockIdx.x * blockDim.x + threadIdx.x) >> 5;
    int lane = threadIdx.x & 31;
    int row0 = wave * 16;
    if (row0 >= M) return;                  // wave-uniform: EXEC stays all-1s

    int half = lane >> 4;                   // 0 or 1
    int l15  = lane & 15;

    v8f acc0 = {}; v8f acc1 = {}; v8f acc2 = {}; v8f acc3 = {};
    v8f acc4 = {}; v8f acc5 = {}; v8f acc6 = {}; v8f acc7 = {};

    const float* xrow = X + (size_t)(row0 + l15) * 128;

#pragma unroll 4
    for (int k = 0; k < 128; k += 4) {
        v2f a;
        a.x = xrow[k + 2 * half + 0];
        a.y = xrow[k + 2 * half + 1];
        const float* wbase = W + (size_t)(k + 2 * half) * 128 + l15;
#define DO_TILE(T, ACC)                                                          \
        {                                                                        \
            v2f b;                                                               \
            b.x = wbase[(T) * 16];                                               \
            b.y = wbase[(T) * 16 + 128];                                         \
            ACC = __builtin_amdgcn_wmma_f32_16x16x4_f32(                         \
                false, a, false, b, (short)0, ACC, false, false);                \
        }
        DO_TILE(0, acc0) DO_TILE(1, acc1) DO_TILE(2, acc2) DO_TILE(3, acc3)
        DO_TILE(4, acc4) DO_TILE(5, acc5) DO_TILE(6, acc6) DO_TILE(7, acc7)
#undef DO_TILE
    }

    float* hbase = H + (size_t)row0 * 128 + l15;
#define ST_TILE(T, ACC)                                                          \
    {                                                                            \
        _Pragma("unroll")                                                        \
        for (int r = 0; r < 8; ++r)                                              \
            hbase[(size_t)(r + 8 * half) * 128 + (T) * 16] = ACC[r];             \
    }
    ST_TILE(0, acc0) ST_TILE(1, acc1) ST_TILE(2, acc2) ST_TILE(3, acc3)
    ST_TILE(4, acc4) ST_TILE(5, acc5) ST_TILE(6, acc6) ST_TILE(7, acc7)
#undef ST_TILE
}

// ---------------------------------------------------------------------------
// Edge scatter: agg[dst] += H[src] * (dis[src]*dis[dst]).  One wave per edge;
// each lane handles a float4 chunk (coalesced gather, L2-resolved atomics).
// ---------------------------------------------------------------------------
__global__ void k_edge_scatter(const float* __restrict__ H,
                               const int* __restrict__ src,
                               const int* __restrict__ dst,
                               const float* __restrict__ dis,
                               float* __restrict__ agg, int E) {
    int wave = (blockIdx.x * blockDim.x + threadIdx.x) >> 5;
    int lane = threadIdx.x & 31;
    if (wave >= E) return;
    int s = src[wave];
    int d = dst[wave];
    float norm = dis[s] * dis[d];
    const float4* hs = (const float4*)(H + (size_t)s * 128);
    float4 v = hs[lane];
    float* out = agg + (size_t)d * 128 + lane * 4;
    atomicAdd(out + 0, v.x * norm);
    atomicAdd(out + 1, v.y * norm);
    atomicAdd(out + 2, v.z * norm);
    atomicAdd(out + 3, v.w * norm);
}

// ---------------------------------------------------------------------------
// out = relu(agg + H * dis^2 + b)
// ---------------------------------------------------------------------------
__global__ void k_self_bias_relu(const float* __restrict__ agg,
                                 const float* __restrict__ H,
                                 const float* __restrict__ dis,
                                 const float* __restrict__ b,
                                 float* __restrict__ out, long total) {
    long i = (long)blockIdx.x * blockDim.x + threadIdx.x;
    if (i >= total) return;
    int node = (int)(i >> 7);
    int f    = (int)(i & 127);
    float di = dis[node];
    float v = agg[i] + H[i] * di * di + b[f];
    out[i] = fmaxf(v, 0.0f);
}

// ---------------------------------------------------------------------------
// Pooling accumulation: one wave per node.
//   gnn_sum[g] += act[node]; gnn_cnt[g] += 1
//   cb_sum[g]  += x[node] * (xsrc==1); cb_cnt[g] += (xsrc==1)
// ---------------------------------------------------------------------------
__global__ void k_pool_acc(const float* __restrict__ act,
                           const float* __restrict__ x,
                           const int* __restrict__ batch,
                           const int* __restrict__ xsrc,
                           float* __restrict__ gnn_sum, float* __restrict__ gnn_cnt,
                           float* __restrict__ cb_sum,  float* __restrict__ cb_cnt,
                           int N) {
    int wave = (blockIdx.x * blockDim.x + threadIdx.x) >> 5;
    int lane = threadIdx.x & 31;
    if (wave >= N) return;
    int g = batch[wave];
    float m = (xsrc[wave] == 1) ? 1.0f : 0.0f;

    const float4* av = (const float4*)(act + (size_t)wave * 128);
    float4 a = av[lane];
    float* gs = gnn_sum + (size_t)g * 128 + lane * 4;
    atomicAdd(gs + 0, a.x); atomicAdd(gs + 1, a.y);
    atomicAdd(gs + 2, a.z); atomicAdd(gs + 3, a.w);

    if (m != 0.0f) {
        const float4* xv = (const float4*)(x + (size_t)wave * 128);
        float4 xx = xv[lane];
        float* cs = cb_sum + (size_t)g * 128 + lane * 4;
        atomicAdd(cs + 0, xx.x); atomicAdd(cs + 1, xx.y);
        atomicAdd(cs + 2, xx.z); atomicAdd(cs + 3, xx.w);
    }
    if (lane == 0) {
        atomicAdd(&gnn_cnt[g], 1.0f);
        atomicAdd(&cb_cnt[g], m);
    }
}

// ---------------------------------------------------------------------------
// Head: out[g,c] = concat(gnn_mean[g], cb_mean[g]) . Wh[:,c] + bh[c]
// ---------------------------------------------------------------------------
__global__ void k_head(const float* __restrict__ gnn_sum, const float* __restrict__ gnn_cnt,
                       const float* __restrict__ cb_sum,  const float* __restrict__ cb_cnt,
                       const float* __restrict__ Wh, const float* __restrict__ bh,
                       float* __restrict__ out, int G) {
    int i = blockIdx.x * blockDim.x + threadIdx.x;
    if (i >= G * 2) return;
    int g = i >> 1;
    int c = i & 1;
    float ig = 1.0f / fmaxf(gnn_cnt[g], 1.0f);
    float ic = 1.0f / fmaxf(cb_cnt[g], 1.0f);
    float acc = bh[c];
    const float* gs = gnn_sum + (size_t)g * 128;
    const float* cs = cb_sum + (size_t)g * 128;
    for (int k = 0; k < 128; ++k) acc += gs[k] * ig * Wh[k * 2 + c];
    for (int k = 0; k < 128; ++k) acc += cs[k] * ic * Wh[(128 + k) * 2 + c];
    out[i] = acc;
}

// ---------------------------------------------------------------------------
extern "C" void kernel_launch(void* const* d_in, const int* in_sizes, int n_in,
                              void* d_out, int out_size, void* d_ws, size_t ws_size,
                              hipStream_t stream) {
    const float* x    = (const float*)d_in[0];
    const int*   ei   = (const int*)d_in[1];
    const int*   batch= (const int*)d_in[2];
    const int*   xsrc = (const int*)d_in[3];
    const float* W1   = (const float*)d_in[4];
    const float* b1   = (const float*)d_in[5];
    const float* W2   = (const float*)d_in[6];
    const float* b2   = (const float*)d_in[7];
    const float* Wh   = (const float*)d_in[8];
    const float* bh   = (const float*)d_in[9];
    float* out = (float*)d_out;

    const int N = in_sizes[2];          // 50000 (batch length)
    const int E = in_sizes[1] / 2;      // 800000
    const int F = 128;
    const int G = 512;
    const int* src = ei;
    const int* dst = ei + E;

    // Workspace carve-up (floats)
    float* ws = (float*)d_ws;
    float* deg = ws;                 ws += N;
    float* dis = ws;                 ws += N;
    float* H   = ws;                 ws += (size_t)N * F;   // GEMM output
    float* AG  = ws;                 ws += (size_t)N * F;   // scatter accumulator
    float* ACT = ws;                 ws += (size_t)N * F;   // activated features
    float* gnn_sum = ws;             ws += (size_t)G * F;
    float* gnn_cnt = ws;             ws += G;
    float* cb_sum  = ws;             ws += (size_t)G * F;
    float* cb_cnt  = ws;             ws += G;
    long poolN = (long)G * F * 2 + 2 * G;
    long NF = (long)N * F;

    const int TB = 256;
    int fillBlocks = 2048;

    // deg = 1 (self loop); dis = rsqrt(deg after scatter)
    k_fill_f32<<<fillBlocks, TB, 0, stream>>>(deg, 1.0f, N);
    k_degree<<<(E + TB - 1) / TB, TB, 0, stream>>>(dst, deg, E);
    k_rsqrt<<<(N + TB - 1) / TB, TB, 0, stream>>>(deg, dis, N);

    int strips = (N + 15) / 16;
    int gemmBlocks = (strips + (TB / 32) - 1) / (TB / 32);
    int edgeBlocks = (E * 32 + TB - 1) / TB;
    int nodeWaveBlocks = (N * 32 + TB - 1) / TB;

    // -------- Layer 1 --------
    k_gemm_wmma_f32<<<gemmBlocks, TB, 0, stream>>>(x, W1, H, N);
    k_fill_f32<<<fillBlocks, TB, 0, stream>>>(AG, 0.0f, NF);
    k_edge_scatter<<<edgeBlocks, TB, 0, stream>>>(H, src, dst, dis, AG, E);
    k_self_bias_relu<<<(int)((NF + TB - 1) / TB), TB, 0, stream>>>(AG, H, dis, b1, ACT, NF);

    // -------- Layer 2 --------
    k_gemm_wmma_f32<<<gemmBlocks, TB, 0, stream>>>(ACT, W2, H, N);
    k_fill_f32<<<fillBlocks, TB, 0, stream>>>(AG, 0.0f, NF);
    k_edge_scatter<<<edgeBlocks, TB, 0, stream>>>(H, src, dst, dis, AG, E);
    k_self_bias_relu<<<(int)((NF + TB - 1) / TB), TB, 0, stream>>>(AG, H, dis, b2, ACT, NF);

    // -------- Pooling + head --------
    k_fill_f32<<<fillBlocks, TB, 0, stream>>>(gnn_sum, 0.0f, poolN);
    k_pool_acc<<<nodeWaveBlocks, TB, 0, stream>>>(ACT, x, batch, xsrc,
                                                  gnn_sum, gnn_cnt, cb_sum, cb_cnt, N);
    k_head<<<(G * 2 + TB - 1) / TB, TB, 0, stream>>>(gnn_sum, gnn_cnt, cb_sum, cb_cnt,
                                                     Wh, bh, out, G);
}